// Uniformity_40699110097720
// MI455X (gfx1250) — compile-verified
//
#include <hip/hip_runtime.h>

#define N_PTS 8192
#define DIMS  256
#define TILE  128                          // BM = BN
#define BK    64                           // K chunk through LDS
#define LDS_STRIDE (BK + 4)                // 68 floats; 68 % 64 banks = 4 -> conflict-free frags
#define CHUNK_FLTS (TILE * LDS_STRIDE)     // one tile-side chunk in LDS
#define NTILES (N_PTS / TILE)              // 64
#define NPAIRS (NTILES * (NTILES + 1) / 2) // 2080 upper-tri tile pairs
#define NCHUNKS (DIMS / BK)                // 4
#define T_COEF 2.0f

typedef __attribute__((ext_vector_type(2))) float v2f;
typedef __attribute__((ext_vector_type(8))) float v8f;
typedef int v4i_vs __attribute__((vector_size(16)));  // matches builtin's V4i spelling

// ---------------------------------------------------------------------------
// CDNA5 async global->LDS copy (ASYNCcnt-tracked, no VGPR round trip).
// Builtin signature (from toolchain): (v4i as1*, v4i as3*, imm offset, imm cpol)
// ---------------------------------------------------------------------------
__device__ __forceinline__ void async_copy_b128(const float* g, float* l) {
#if defined(__HIP_DEVICE_COMPILE__)
#if __has_builtin(__builtin_amdgcn_global_load_async_to_lds_b128)
  __builtin_amdgcn_global_load_async_to_lds_b128(
      (__attribute__((address_space(1))) v4i_vs*)g,
      (__attribute__((address_space(3))) v4i_vs*)l, 0, 0);
#else
  unsigned lds_off = (unsigned)(__attribute__((address_space(3))) void*)l;
  asm volatile("global_load_async_to_lds_b128 %0, %1, off"
               :: "v"(lds_off), "v"(g) : "memory");
#endif
#else
  (void)g; (void)l;
#endif
}

template <int N>
__device__ __forceinline__ void wait_asynccnt() {
#if defined(__HIP_DEVICE_COMPILE__)
#if __has_builtin(__builtin_amdgcn_s_wait_asynccnt)
  __builtin_amdgcn_s_wait_asynccnt(N);
#else
  asm volatile("s_wait_asynccnt %0" :: "n"(N) : "memory");
#endif
#endif
}

// ---------------------------------------------------------------------------
// Squared row norms: one wave (32 lanes) per row, shuffle reduce.
// ---------------------------------------------------------------------------
__global__ void norms_kernel(const float* __restrict__ x1,
                             const float* __restrict__ x2,
                             float* __restrict__ norms) {
  const int row = blockIdx.x;
  const int mat = blockIdx.y;
  const float* __restrict__ x = mat ? x2 : x1;
  const int lane = threadIdx.x;  // 0..31
  float s = 0.f;
#pragma unroll
  for (int i = 0; i < DIMS / 32; ++i) {
    float v = x[(size_t)row * DIMS + lane + i * 32];
    s += v * v;
  }
#pragma unroll
  for (int o = 16; o > 0; o >>= 1) s += __shfl_down(s, o, 32);
  if (lane == 0) norms[mat * N_PTS + row] = s;
}

// ---------------------------------------------------------------------------
// Issue the async copies for one 64-wide K chunk of both tile sides.
// 16 instructions per thread (8 per side), b128 each.
// ---------------------------------------------------------------------------
__device__ __forceinline__ void stage_chunk(const float* __restrict__ x,
                                            int rowA, int rowB, int kc,
                                            float* As, float* Bs, int tid) {
#pragma unroll
  for (int i = 0; i < 8; ++i) {
    int idx = tid + i * 256;          // 0..2047
    int r   = idx >> 4;               // 0..127
    int c4  = (idx & 15) << 2;        // 0..60 step 4
    async_copy_b128(&x[(size_t)(rowA + r) * DIMS + kc + c4], &As[r * LDS_STRIDE + c4]);
    async_copy_b128(&x[(size_t)(rowB + r) * DIMS + kc + c4], &Bs[r * LDS_STRIDE + c4]);
  }
}

// ---------------------------------------------------------------------------
// Fused Gram-tile + exp + reduce. One block per upper-triangular 128x128 tile
// pair; 8 waves, each wave owns a 16x128 strip (8 WMMA accumulators).
// Double-buffered async staging hides copy latency behind WMMA.
// ---------------------------------------------------------------------------
__global__ void __launch_bounds__(256)
tile_kernel(const float* __restrict__ x1, const float* __restrict__ x2,
            const float* __restrict__ norms, float* __restrict__ partials) {
  extern __shared__ float smem[];
  // buffer b: A at b*2*CHUNK, B at b*2*CHUNK + CHUNK
  float* bufA[2] = {smem, smem + 2 * CHUNK_FLTS};
  float* bufB[2] = {smem + CHUNK_FLTS, smem + 3 * CHUNK_FLTS};

  const int mat = blockIdx.y;
  const float* __restrict__ x = mat ? x2 : x1;
  const float* __restrict__ nrm = norms + mat * N_PTS;

  // linear pair index -> (ti, tj), ti <= tj
  int p = blockIdx.x;
  int ti = 0;
  while (p >= NTILES - ti) { p -= NTILES - ti; ++ti; }
  const int tj = ti + p;
  const int rowA = ti * TILE, rowB = tj * TILE;

  const int tid  = threadIdx.x;
  const int lane = tid & 31;
  const int wave = tid >> 5;            // 0..7 : subtile row within 128x128 tile
  const int mrow = lane & 15;           // A/B fragment row within 16x16
  const int khi  = (lane >> 4) << 1;    // lanes 16-31 hold K+2, K+3
  const int hiHalf = (lane >> 4) << 3;  // C layout: lanes 16-31 hold M = e+8

  v8f acc[8];
#pragma unroll
  for (int sc = 0; sc < 8; ++sc) acc[sc] = (v8f)(0.f);

  // Prologue: stage chunk 0 into buffer 0.
  stage_chunk(x, rowA, rowB, 0, bufA[0], bufB[0], tid);

#pragma unroll
  for (int c = 0; c < NCHUNKS; ++c) {
    const int cur = c & 1;
    // Overlap: issue next chunk's copies into the other buffer (safe: it was
    // fully consumed before the barrier that ended iteration c-1).
    if (c + 1 < NCHUNKS)
      stage_chunk(x, rowA, rowB, (c + 1) * BK, bufA[1 - cur], bufB[1 - cur], tid);

    // Per-wave ASYNCcnt is in-order: <=16 outstanding => chunk c has landed.
    if (c + 1 < NCHUNKS) wait_asynccnt<16>(); else wait_asynccnt<0>();
    __syncthreads();

    const float* As = bufA[cur];
    const float* Bs = bufB[cur];
#pragma unroll
    for (int ks = 0; ks < BK; ks += 4) {
      v2f a = *(const v2f*)&As[(wave * 16 + mrow) * LDS_STRIDE + ks + khi];
#pragma unroll
      for (int sc = 0; sc < 8; ++sc) {
        v2f b = *(const v2f*)&Bs[(sc * 16 + mrow) * LDS_STRIDE + ks + khi];
        // D = A(16x4) * B(4x16) + C  ; f32 WMMA, accumulate over K
        acc[sc] = __builtin_amdgcn_wmma_f32_16x16x4_f32(
            false, a, false, b, (short)0, acc[sc], false, false);
      }
    }
    __syncthreads();  // all waves done reading before buffer is overwritten
  }

  // Epilogue: d^2 = |xm|^2 + |xn|^2 - 2*gram ; sum w * exp(-t * max(d^2, 0)).
  const float wgt = (ti == tj) ? 1.0f : 2.0f;  // off-diagonal tiles counted twice
  float lsum = 0.f;
#pragma unroll
  for (int sc = 0; sc < 8; ++sc) {
    const int n = rowB + sc * 16 + mrow;
    const float nN = nrm[n];
#pragma unroll
    for (int e = 0; e < 8; ++e) {
      const int m = rowA + wave * 16 + hiHalf + e;
      float sq = nrm[m] + nN - 2.0f * acc[sc][e];
      sq = fmaxf(sq, 0.0f);
      lsum += wgt * __expf(-T_COEF * sq);
    }
  }

  // Deterministic block tree-reduce (reuse LDS), one partial per block.
  __syncthreads();
  smem[tid] = lsum;
  __syncthreads();
#pragma unroll
  for (int s = 128; s > 0; s >>= 1) {
    if (tid < s) smem[tid] += smem[tid + s];
    __syncthreads();
  }
  if (tid == 0) partials[mat * NPAIRS + blockIdx.x] = smem[0];
}

// ---------------------------------------------------------------------------
// Deterministic final reduction + log-mean combine.
// ---------------------------------------------------------------------------
__global__ void finalize_kernel(const float* __restrict__ partials,
                                float* __restrict__ out) {
  __shared__ float red[256];
  __shared__ float ures[2];
  const int tid = threadIdx.x;
  for (int mat = 0; mat < 2; ++mat) {
    float s = 0.f;
    for (int pz = tid; pz < NPAIRS; pz += 256) s += partials[mat * NPAIRS + pz];
    red[tid] = s;
    __syncthreads();
    for (int st = 128; st > 0; st >>= 1) {
      if (tid < st) red[tid] += red[tid + st];
      __syncthreads();
    }
    if (tid == 0) {
      const float nf = (float)N_PTS;
      float mean_pairs = (red[0] - nf) / (nf * (nf - 1.0f));
      ures[mat] = __logf(mean_pairs);
    }
    __syncthreads();
  }
  if (tid == 0) out[0] = 0.5f * (ures[0] + ures[1]);
}

// ---------------------------------------------------------------------------
extern "C" void kernel_launch(void* const* d_in, const int* in_sizes, int n_in,
                              void* d_out, int out_size, void* d_ws, size_t ws_size,
                              hipStream_t stream) {
  (void)in_sizes; (void)n_in; (void)out_size; (void)ws_size;
  const float* x1 = (const float*)d_in[0];
  const float* x2 = (const float*)d_in[1];
  float* out      = (float*)d_out;
  float* norms    = (float*)d_ws;           // 2 * 8192 floats
  float* partials = norms + 2 * N_PTS;      // 2 * NPAIRS floats

  norms_kernel<<<dim3(N_PTS, 2), 32, 0, stream>>>(x1, x2, norms);

  const size_t smem_bytes = (size_t)(4 * CHUNK_FLTS) * sizeof(float); // 139,264 B
  tile_kernel<<<dim3(NPAIRS, 2), 256, smem_bytes, stream>>>(x1, x2, norms, partials);

  finalize_kernel<<<1, 256, 0, stream>>>(partials, out);
}